// CRF_34918084116868
// MI455X (gfx1250) — compile-verified
//
#include <hip/hip_runtime.h>

// ---------------------------------------------------------------------------
// CRF forward (log-partition) for B=128, T=1024, F=128, K=32 on gfx1250.
//   Phase 1: emission = input @ W_emit via V_WMMA_F32_16X16X4_F32 (fp32 WMMA)
//   Phase 2: per-batch sequential scan, one wave32 per batch (K==32 lanes)
//   Phase 3: deterministic fixed-order sum of 128 per-batch log_z values
// ---------------------------------------------------------------------------

#define BB 128
#define TT 1024
#define FF 128
#define KK 32

typedef __attribute__((ext_vector_type(2))) float v2f;
typedef __attribute__((ext_vector_type(8))) float v8f;

// ---------------- Phase 1: emission GEMM (M=B*T, K=F, N=KK) ----------------
// Each wave produces a 16x32 strip: two 16x16 f32 accumulators, K-loop of
// 128/4 = 32 WMMA steps per tile.
__global__ __launch_bounds__(256) void crf_emit_gemm(
    const float* __restrict__ A,   // [B*T, 128] row-major
    const float* __restrict__ W,   // [128, 32]  row-major
    float* __restrict__ E)         // [B*T, 32]  row-major
{
    const int lane = threadIdx.x & 31;
    const int wave = threadIdx.x >> 5;
    const long m0  = ((long)blockIdx.x * 8 + wave) * 16;   // row tile base
    const int lm   = lane & 15;   // A: row-in-tile ; B: column
    const int kh   = lane >> 4;   // K half select (0 -> K+{0,1}, 1 -> K+{2,3})

    v8f c0 = {};   // columns 0..15
    v8f c1 = {};   // columns 16..31

    const float* arow = A + (m0 + lm) * FF;

#pragma unroll 4
    for (int k0 = 0; k0 < FF; k0 += 4) {
        const int ka = k0 + kh * 2;
        // A fragment 16x4 f32: lane holds row lm, K = ka, ka+1 (contiguous b64)
        v2f a = *(const v2f*)(arow + ka);
        // B fragment 4x16 f32: lane holds column lm (and lm+16 for tile 1)
        v2f b0, b1;
        b0.x = W[(ka + 0) * KK + lm];
        b0.y = W[(ka + 1) * KK + lm];
        b1.x = W[(ka + 0) * KK + 16 + lm];
        b1.y = W[(ka + 1) * KK + 16 + lm];
        c0 = __builtin_amdgcn_wmma_f32_16x16x4_f32(
                 false, a, false, b0, (short)0, c0, false, false);
        c1 = __builtin_amdgcn_wmma_f32_16x16x4_f32(
                 false, a, false, b1, (short)0, c1, false, false);
    }

    // C/D layout: lanes 0-15 -> N=lane, M=r ; lanes 16-31 -> N=lane-16, M=r+8
    float* erow = E + (m0 + (long)kh * 8) * KK + lm;
#pragma unroll
    for (int r = 0; r < 8; ++r) {
        erow[(long)r * KK]      = c0[r];
        erow[(long)r * KK + 16] = c1[r];
    }
}

// ---------------- Phase 2: sequential scan, one wave per batch -------------
__device__ __forceinline__ float lane_bcast(float x, int l) {
    return __int_as_float(__builtin_amdgcn_readlane(__float_as_int(x), l));
}

__device__ __forceinline__ float wave_max32(float m) {
    m = fmaxf(m, __shfl_xor(m, 16, 32));
    m = fmaxf(m, __shfl_xor(m,  8, 32));
    m = fmaxf(m, __shfl_xor(m,  4, 32));
    m = fmaxf(m, __shfl_xor(m,  2, 32));
    m = fmaxf(m, __shfl_xor(m,  1, 32));
    return m;
}

__device__ __forceinline__ float wave_sum32(float s) {
    s += __shfl_xor(s, 16, 32);
    s += __shfl_xor(s,  8, 32);
    s += __shfl_xor(s,  4, 32);
    s += __shfl_xor(s,  2, 32);
    s += __shfl_xor(s,  1, 32);
    return s;
}

__global__ __launch_bounds__(32) void crf_scan(
    const float* __restrict__ Em,           // [B, T, K]
    const unsigned char* __restrict__ mask, // [B, T] (jax bool -> i8)
    const float* __restrict__ trans,        // [K, K]
    float* __restrict__ partial)            // [B]
{
    const int b = blockIdx.x;
    const int k = threadIdx.x;              // lane == state index, K == 32

    // Pre-exponentiated transition column: Ecol[j] = exp(trans[j][k]).
    // Factoring a shared max m out of logsumexp_j turns each scan step into
    // a 32-MAC dot product: new_k = m + log( sum_j exp(a_j-m)*Ecol[j] ) + em.
    float Ecol[KK];
#pragma unroll
    for (int j = 0; j < KK; ++j) Ecol[j] = __expf(trans[j * KK + k]);

    const float* em = Em + (long)b * TT * KK;
    const unsigned char* mk = mask + (long)b * TT;

    float a = em[k];                        // alpha0 = emission[:,0]

    for (int t = 1; t < TT; ++t) {
        const float e = em[(long)t * KK + k];
        const unsigned char mt = mk[t];

        const float m = wave_max32(a);
        const float v = __expf(a - m);      // lane j holds exp(alpha_j - m)

        float s = 0.0f;
#pragma unroll
        for (int j = 0; j < KK; ++j)
            s = fmaf(lane_bcast(v, j), Ecol[j], s);

        const float anew = m + __logf(s) + e;
        a = mt ? anew : a;
    }

    // log_z_b = logsumexp_k(alpha)
    const float m = wave_max32(a);
    const float s = wave_sum32(__expf(a - m));
    if (k == 0) partial[b] = m + __logf(s);
}

// ---------------- Phase 3: deterministic final reduction -------------------
__global__ void crf_final_sum(const float* __restrict__ partial,
                              float* __restrict__ out)
{
    if (threadIdx.x == 0 && blockIdx.x == 0) {
        float s = 0.0f;
        for (int i = 0; i < BB; ++i) s += partial[i];
        out[0] = s;
    }
}

// ---------------------------------------------------------------------------
extern "C" void kernel_launch(void* const* d_in, const int* in_sizes, int n_in,
                              void* d_out, int out_size, void* d_ws, size_t ws_size,
                              hipStream_t stream)
{
    const float*         inp   = (const float*)d_in[0];          // (B,T,F) f32
    const unsigned char* mask  = (const unsigned char*)d_in[1];  // (B,T) bool
    const float*         W     = (const float*)d_in[2];          // (F,K) f32
    const float*         trans = (const float*)d_in[3];          // (K,K) f32

    float* ws      = (float*)d_ws;
    float* Em      = ws;                             // B*T*K floats (16 MB)
    float* partial = ws + (size_t)BB * TT * KK;      // B floats

    // Phase 1: (B*T) rows / (8 waves * 16 rows) = 1024 blocks of 256 threads
    crf_emit_gemm<<<(BB * TT) / (8 * 16), 256, 0, stream>>>(inp, W, Em);

    // Phase 2: one wave32 per batch element
    crf_scan<<<BB, KK, 0, stream>>>(Em, mask, trans, partial);

    // Phase 3: fixed-order scalar sum (deterministic)
    crf_final_sum<<<1, 1, 0, stream>>>(partial, (float*)d_out);
}